// MambaBlock_50096498540839
// MI455X (gfx1250) — compile-verified
//
#include <hip/hip_runtime.h>
#include <math.h>

#define D_MODEL 1024
#define D_STATE 16
#define D_CONV  4
#define D_INNER 2048
#define DT_RANK 64
#define XDBL_W  (DT_RANK + 2 * D_STATE)   // 96

typedef __attribute__((ext_vector_type(2))) float v2f;
typedef __attribute__((ext_vector_type(8))) float v8f;

// ---------------------------------------------------------------------------
// GEMM: C[M,N] = epi( A[M x K] (row stride lda) @ W[N x K]^T )
// epi 0: identity; epi 1: softplus(x + bias[n])   (dt_proj)
// One wave computes a 32x32 tile via V_WMMA_F32_16X16X4_F32; 4 waves/block
// cover a 64x64 block tile. K consumed 16 at a time with v8f (b128) loads;
// K-order inside the WMMA is permuted identically for A and W, which leaves
// the dot product unchanged.
// ---------------------------------------------------------------------------
__global__ __launch_bounds__(128)
void gemm_f32_wmma(const float* __restrict__ A, const float* __restrict__ W,
                   const float* __restrict__ bias, float* __restrict__ C,
                   int M, int N, int K, int lda, int ldc, int epi)
{
    const int lane = threadIdx.x & 31;
    const int wave = threadIdx.x >> 5;
    const int half = lane >> 4;       // K-half selector within a fragment
    const int r    = lane & 15;       // row (A) / row-of-W (B) index

    const int m0 = blockIdx.y * 64 + (wave >> 1) * 32;
    const int n0 = blockIdx.x * 64 + (wave & 1) * 32;

    // clamp out-of-range fragment rows (stores are guarded; values unused)
    const int ra0 = (m0 + r      < M) ? (m0 + r)      : (M - 1);
    const int ra1 = (m0 + 16 + r < M) ? (m0 + 16 + r) : (M - 1);
    const int rw0 = (n0 + r      < N) ? (n0 + r)      : (N - 1);
    const int rw1 = (n0 + 16 + r < N) ? (n0 + 16 + r) : (N - 1);

    const float* pa0 = A + (size_t)ra0 * lda + 8 * half;
    const float* pa1 = A + (size_t)ra1 * lda + 8 * half;
    const float* pw0 = W + (size_t)rw0 * K   + 8 * half;
    const float* pw1 = W + (size_t)rw1 * K   + 8 * half;

    v8f acc00 = {}; v8f acc01 = {}; v8f acc10 = {}; v8f acc11 = {};

    for (int k0 = 0; k0 < K; k0 += 16) {
        v8f a0 = *(const v8f*)(pa0 + k0);
        v8f a1 = *(const v8f*)(pa1 + k0);
        v8f w0 = *(const v8f*)(pw0 + k0);
        v8f w1 = *(const v8f*)(pw1 + k0);
#pragma unroll
        for (int s = 0; s < 4; ++s) {
            v2f fa0 = { a0[2 * s], a0[2 * s + 1] };
            v2f fa1 = { a1[2 * s], a1[2 * s + 1] };
            v2f fb0 = { w0[2 * s], w0[2 * s + 1] };
            v2f fb1 = { w1[2 * s], w1[2 * s + 1] };
            acc00 = __builtin_amdgcn_wmma_f32_16x16x4_f32(false, fa0, false, fb0,
                                                          (short)0, acc00, false, false);
            acc01 = __builtin_amdgcn_wmma_f32_16x16x4_f32(false, fa0, false, fb1,
                                                          (short)0, acc01, false, false);
            acc10 = __builtin_amdgcn_wmma_f32_16x16x4_f32(false, fa1, false, fb0,
                                                          (short)0, acc10, false, false);
            acc11 = __builtin_amdgcn_wmma_f32_16x16x4_f32(false, fa1, false, fb1,
                                                          (short)0, acc11, false, false);
        }
    }

    const v8f accs[2][2] = { { acc00, acc01 }, { acc10, acc11 } };
#pragma unroll
    for (int i = 0; i < 2; ++i) {
#pragma unroll
        for (int j = 0; j < 2; ++j) {
            const int col = n0 + 16 * j + r;
            if (col >= N) continue;
            float b = (epi == 1) ? bias[col] : 0.0f;
#pragma unroll
            for (int v = 0; v < 8; ++v) {
                const int row = m0 + 16 * i + 8 * half + v;   // C/D lane layout
                if (row >= M) continue;
                float val = accs[i][j][v];
                if (epi == 1) {
                    val += b;
                    val = (val > 20.0f) ? val : log1pf(__expf(val)); // softplus
                }
                C[(size_t)row * ldc + col] = val;
            }
        }
    }
}

// ---------------------------------------------------------------------------
// Causal depthwise conv1d (W=4) + bias + SiLU.
// Reads the x-half of xz ([BL, 4096], channels 0..2047), writes xc [BL, 2048].
// ---------------------------------------------------------------------------
__global__ __launch_bounds__(256)
void conv_silu_kernel(const float* __restrict__ xz, const float* __restrict__ cw,
                      const float* __restrict__ cb, float* __restrict__ xc,
                      int L, long total)
{
    long idx = (long)blockIdx.x * blockDim.x + threadIdx.x;
    if (idx >= total) return;
    int d  = (int)(idx % D_INNER);
    long bl = idx / D_INNER;            // b * L + l
    int l  = (int)(bl % L);
    long b = bl / L;

    float acc = cb[d];
#pragma unroll
    for (int w = 0; w < D_CONV; ++w) {
        int ll = l - (D_CONV - 1) + w;
        if (ll >= 0)
            acc += cw[d * D_CONV + w] * xz[(b * L + ll) * (size_t)(2 * D_INNER) + d];
    }
    float sig = 1.0f / (1.0f + __expf(-acc));
    xc[bl * (size_t)D_INNER + d] = acc * sig;   // SiLU
}

// ---------------------------------------------------------------------------
// Selective scan + skip + gating.
// One thread per (b, channel d). h[16] in registers. B_t / C_t staged in LDS.
// Writes g = (y + D*x) * silu(z)   [BL, 2048]
// ---------------------------------------------------------------------------
__global__ __launch_bounds__(256)
void scan_kernel(const float* __restrict__ xdbl, const float* __restrict__ delta,
                 const float* __restrict__ xc,   const float* __restrict__ xz,
                 const float* __restrict__ A_log, const float* __restrict__ Dw,
                 float* __restrict__ g, int L)
{
    const int d = blockIdx.x * blockDim.x + threadIdx.x;
    const int b = blockIdx.y;

    float negA[D_STATE];
    float h[D_STATE];
#pragma unroll
    for (int n = 0; n < D_STATE; ++n) {
        negA[n] = -__expf(A_log[(size_t)d * D_STATE + n]);
        h[n] = 0.0f;
    }
    const float Dd = Dw[d];

    __shared__ float sBC[2 * D_STATE];   // [0..15]=B_t, [16..31]=C_t

    for (int l = 0; l < L; ++l) {
        const size_t row = (size_t)b * L + l;
        if (threadIdx.x < 2 * D_STATE)
            sBC[threadIdx.x] = xdbl[row * XDBL_W + DT_RANK + threadIdx.x];
        __syncthreads();

        const float dt = delta[row * D_INNER + d];
        const float xv = xc[row * D_INNER + d];
        float y = 0.0f;
#pragma unroll
        for (int n = 0; n < D_STATE; ++n) {
            const float dA = __expf(dt * negA[n]);
            h[n] = dA * h[n] + (dt * sBC[n]) * xv;
            y += h[n] * sBC[D_STATE + n];
        }
        const float z = xz[row * (size_t)(2 * D_INNER) + D_INNER + d];
        const float gate = z / (1.0f + __expf(-z));     // silu(z)
        g[row * D_INNER + d] = (y + Dd * xv) * gate;
        __syncthreads();
    }
}

// ---------------------------------------------------------------------------
extern "C" void kernel_launch(void* const* d_in, const int* in_sizes, int n_in,
                              void* d_out, int out_size, void* d_ws, size_t ws_size,
                              hipStream_t stream)
{
    const float* x         = (const float*)d_in[0];
    const float* in_proj_w = (const float*)d_in[1];
    const float* conv_w    = (const float*)d_in[2];
    const float* conv_b    = (const float*)d_in[3];
    const float* x_proj_w  = (const float*)d_in[4];
    const float* dt_proj_w = (const float*)d_in[5];
    const float* dt_proj_b = (const float*)d_in[6];
    const float* A_log     = (const float*)d_in[7];
    const float* Dw        = (const float*)d_in[8];
    const float* out_projw = (const float*)d_in[9];
    (void)n_in; (void)ws_size;

    const int BL = in_sizes[0] / D_MODEL;   // B * L = 4096
    const int B  = 2;
    const int L  = BL / B;

    // workspace layout (all offsets multiples of 4096 B for b128 alignment)
    char* ws = (char*)d_ws;
    float* xz    = (float*)ws;                                        // [BL, 4096]
    size_t off   = (size_t)BL * 2 * D_INNER * sizeof(float);
    float* xc    = (float*)(ws + off);                                // [BL, 2048]
    off         += (size_t)BL * D_INNER * sizeof(float);
    float* xdbl  = (float*)(ws + off);                                // [BL, 96]
    off         += (size_t)BL * XDBL_W * sizeof(float);
    float* delta = (float*)(ws + off);                                // [BL, 2048]
    off         += (size_t)BL * D_INNER * sizeof(float);
    float* g     = (float*)(ws + off);                                // [BL, 2048]

    // 1) in_proj: xz = x @ in_proj_w^T        [BL,1024] x [4096,1024]^T
    {
        dim3 grid((2 * D_INNER + 63) / 64, (BL + 63) / 64);
        gemm_f32_wmma<<<grid, 128, 0, stream>>>(x, in_proj_w, nullptr, xz,
                                                BL, 2 * D_INNER, D_MODEL,
                                                D_MODEL, 2 * D_INNER, 0);
    }
    // 2) causal depthwise conv + SiLU
    {
        long total = (long)BL * D_INNER;
        int blocks = (int)((total + 255) / 256);
        conv_silu_kernel<<<blocks, 256, 0, stream>>>(xz, conv_w, conv_b, xc, L, total);
    }
    // 3) x_proj: xdbl = xc @ x_proj_w^T       [BL,2048] x [96,2048]^T
    {
        dim3 grid((XDBL_W + 63) / 64, (BL + 63) / 64);
        gemm_f32_wmma<<<grid, 128, 0, stream>>>(xc, x_proj_w, nullptr, xdbl,
                                                BL, XDBL_W, D_INNER,
                                                D_INNER, XDBL_W, 0);
    }
    // 4) dt_proj: delta = softplus(xdbl[:, :64] @ dt_proj_w^T + b)
    {
        dim3 grid((D_INNER + 63) / 64, (BL + 63) / 64);
        gemm_f32_wmma<<<grid, 128, 0, stream>>>(xdbl, dt_proj_w, dt_proj_b, delta,
                                                BL, D_INNER, DT_RANK,
                                                XDBL_W, D_INNER, 1);
    }
    // 5) selective scan + skip + gating -> g
    {
        dim3 grid(D_INNER / 256, B);
        scan_kernel<<<grid, 256, 0, stream>>>(xdbl, delta, xc, xz, A_log, Dw, g, L);
    }
    // 6) out_proj: out = g @ out_proj_w^T     [BL,2048] x [1024,2048]^T
    {
        dim3 grid((D_MODEL + 63) / 64, (BL + 63) / 64);
        gemm_f32_wmma<<<grid, 128, 0, stream>>>(g, out_projw, nullptr, (float*)d_out,
                                                BL, D_MODEL, D_INNER,
                                                D_INNER, D_MODEL, 0);
    }
    (void)out_size;
}